// KeyFeatureFusion_7834020348449
// MI455X (gfx1250) — compile-verified
//
#include <hip/hip_runtime.h>

// Problem constants (match reference setup_inputs)
constexpr int kB   = 2;
constexpr int kN   = 8192;
constexpr int kKK  = 1024;
constexpr int kC   = 128;
constexpr int kKNN = 20;
constexpr int kM   = kB * kKK;   // 2048 key points total

typedef __attribute__((ext_vector_type(2))) float v2f;
typedef __attribute__((ext_vector_type(8))) float v8f;

// ---------------------------------------------------------------------------
// Kernel 1: KNN for the 1024 key queries per batch only (topidx rows).
// 1 wave (32 lanes) per query, 8 queries per 256-thread block.
// Block stages 1024 points into LDS; each lane keeps a sorted register top-20;
// a 32-way shuffle merge pops the global top-20 in ascending distance.
// ---------------------------------------------------------------------------
__global__ __launch_bounds__(256) void knn_kernel(
    const float* __restrict__ refinepoint,  // [B,N,3]
    const int*   __restrict__ topidx,       // [B,KK]
    int*         __restrict__ knn_idx)      // [B*KK, 20]
{
  __shared__ float sx[1024], sy[1024], sz[1024];
  const int lane = threadIdx.x & 31;
  const int wave = threadIdx.x >> 5;
  const int q    = blockIdx.x * 8 + wave;   // 0..2047 (blocks never straddle b)
  const int b    = q >> 10;                 // kKK == 1024
  const int qi   = topidx[q];
  const float qx = refinepoint[(b * kN + qi) * 3 + 0];
  const float qy = refinepoint[(b * kN + qi) * 3 + 1];
  const float qz = refinepoint[(b * kN + qi) * 3 + 2];

  float bd[kKNN]; int bi[kKNN];
#pragma unroll
  for (int s = 0; s < kKNN; ++s) { bd[s] = 3.4e38f; bi[s] = 0x7fffffff; }

  for (int tile = 0; tile < kN; tile += 1024) {
    __syncthreads();
    for (int i = threadIdx.x; i < 1024; i += 256) {
      const int g = (b * kN + tile + i) * 3;
      sx[i] = refinepoint[g + 0];
      sy[i] = refinepoint[g + 1];
      sz[i] = refinepoint[g + 2];
    }
    __syncthreads();
    for (int j = lane; j < 1024; j += 32) {
      const float dx = sx[j] - qx, dy = sy[j] - qy, dz = sz[j] - qz;
      const float d  = dx * dx + dy * dy + dz * dz;
      const int gid  = tile + j;
      if (d < bd[kKNN - 1] || (d == bd[kKNN - 1] && gid < bi[kKNN - 1])) {
        bd[kKNN - 1] = d; bi[kKNN - 1] = gid;
#pragma unroll
        for (int s = kKNN - 1; s > 0; --s) {
          const bool sw = (bd[s] < bd[s - 1]) ||
                          (bd[s] == bd[s - 1] && bi[s] < bi[s - 1]);
          if (sw) {
            float td = bd[s]; bd[s] = bd[s - 1]; bd[s - 1] = td;
            int   ti = bi[s]; bi[s] = bi[s - 1]; bi[s - 1] = ti;
          }
        }
      }
    }
  }

  // Merge 32 per-lane sorted lists -> global 20 nearest (wave32 shuffles).
  for (int r = 0; r < kKNN; ++r) {
    float v = bd[0]; int pid = bi[0];
#pragma unroll
    for (int off = 16; off >= 1; off >>= 1) {
      const float ov = __shfl_xor(v, off, 32);
      const int   op = __shfl_xor(pid, off, 32);
      if (ov < v || (ov == v && op < pid)) { v = ov; pid = op; }
    }
    if (lane == 0) knn_idx[q * kKNN + r] = pid;
    if (bd[0] == v && bi[0] == pid) {   // unique winner lane pops its head
#pragma unroll
      for (int s = 0; s < kKNN - 1; ++s) { bd[s] = bd[s + 1]; bi[s] = bi[s + 1]; }
      bd[kKNN - 1] = 3.4e38f; bi[kKNN - 1] = 0x7fffffff;
    }
  }
}

// ---------------------------------------------------------------------------
// Kernel 2: feature[b,kk,c] = keyfeature[b,kk,c]
//                           + mean_j( allfeature[b, idx_j, c] * weight[b, idx_j] )
// One 128-thread block per key point (thread == channel).
// ---------------------------------------------------------------------------
__global__ __launch_bounds__(128) void feat_kernel(
    const float* __restrict__ allfeature,   // [B,N,C]
    const float* __restrict__ weight,       // [B,N]
    const float* __restrict__ keyfeature,   // [B,KK,C]
    const int*   __restrict__ knn_idx,      // [B*KK, 20]
    float*       __restrict__ feature)      // [B*KK, C]
{
  const int p = blockIdx.x;        // b*KK + kk
  const int c = threadIdx.x;       // 0..127
  const int b = p >> 10;
  float acc = 0.f;
#pragma unroll
  for (int j = 0; j < kKNN; ++j) {
    const int nb = knn_idx[p * kKNN + j];
    acc += allfeature[(b * kN + nb) * kC + c] * weight[b * kN + nb];
  }
  feature[p * kC + c] = keyfeature[p * kC + c] + acc * (1.f / (float)kKNN);
}

// ---------------------------------------------------------------------------
// Kernel 3: y[b,o,m] = sum_c conv_w[o,c] * feature[b,m,c] + conv_b[o]
// via V_WMMA_F32_16X16X4_F32 (fp32 in, fp32 acc — matches reference precision).
// One wave per 16(o) x 16(m) tile, K=128 in 32 steps of 4.
// A 16x4 layout: lane l -> row o0+l%16, K = 2*(l/16)+v (v = vgpr 0/1)
// B 4x16 layout: lane l -> col m0+l%16, K = 2*(l/16)+v
// D 16x16 layout: vgpr r, lane l -> o = o0 + r + 8*(l/16), m = m0 + l%16
// ---------------------------------------------------------------------------
__global__ __launch_bounds__(128) void gemm_kernel(
    const float* __restrict__ conv_w,   // [C,C] row-major [o][c]
    const float* __restrict__ conv_b,   // [C]
    const float* __restrict__ feature,  // [B*KK, C]
    float*       __restrict__ y)        // [B,C,KK]
{
  const int tile = blockIdx.x * 4 + (threadIdx.x >> 5);  // 0..1023
  const int lane = threadIdx.x & 31;
  const int bB   = tile >> 9;           // 512 tiles per batch
  const int t    = tile & 511;
  const int o0   = (t >> 6) * 16;       // 8 o-tiles
  const int m0   = (t & 63) * 16;       // 64 m-tiles
  const int lm   = lane & 15, lh = lane >> 4;

  const float* Wp = conv_w  + (o0 + lm) * kC + lh * 2;
  const float* Fp = feature + (bB * kKK + m0 + lm) * kC + lh * 2;

  v8f acc = {};
  for (int k0 = 0; k0 < kC; k0 += 4) {
    v2f a;  a.x  = Wp[k0]; a.y  = Wp[k0 + 1];
    v2f bv; bv.x = Fp[k0]; bv.y = Fp[k0 + 1];
    acc = __builtin_amdgcn_wmma_f32_16x16x4_f32(
        /*neg_a=*/false, a, /*neg_b=*/false, bv,
        /*c_mod=*/(short)0, acc, /*reuse_a=*/false, /*reuse_b=*/false);
  }
#pragma unroll
  for (int r = 0; r < 8; ++r) {
    const int o = o0 + r + lh * 8;
    const int m = m0 + lm;
    y[(bB * kC + o) * kKK + m] = acc[r] + conv_b[o];
  }
}

// ---------------------------------------------------------------------------
// Kernel 4: per-channel batch stats over (B, KK) -> fused scale/shift.
// One 256-thread block per channel.
// ---------------------------------------------------------------------------
__global__ __launch_bounds__(256) void stats_kernel(
    const float* __restrict__ y,       // [B,C,KK]
    const float* __restrict__ gamma,
    const float* __restrict__ beta,
    float*       __restrict__ stats)   // [C][2] = {scale, shift}
{
  __shared__ float ss[8], ss2[8];
  const int o = blockIdx.x;
  float s = 0.f, s2 = 0.f;
  for (int i = threadIdx.x; i < kM; i += 256) {
    const int bb = i >> 10, m = i & 1023;
    const float v = y[(bb * kC + o) * kKK + m];
    s += v; s2 += v * v;
  }
#pragma unroll
  for (int off = 16; off >= 1; off >>= 1) {
    s  += __shfl_xor(s, off, 32);
    s2 += __shfl_xor(s2, off, 32);
  }
  const int lane = threadIdx.x & 31, wave = threadIdx.x >> 5;
  if (lane == 0) { ss[wave] = s; ss2[wave] = s2; }
  __syncthreads();
  if (threadIdx.x == 0) {
    float ts = 0.f, ts2 = 0.f;
#pragma unroll
    for (int w = 0; w < 8; ++w) { ts += ss[w]; ts2 += ss2[w]; }
    const float inv  = 1.f / (float)kM;
    const float mean = ts * inv;
    const float var  = ts2 * inv - mean * mean;   // population var (jnp.var)
    const float sc   = gamma[o] * rsqrtf(var + 1e-5f);
    stats[2 * o]     = sc;
    stats[2 * o + 1] = beta[o] - mean * sc;
  }
}

// ---------------------------------------------------------------------------
// Kernel 5: in-place BN apply + LeakyReLU(0.2).
// ---------------------------------------------------------------------------
__global__ __launch_bounds__(256) void bn_lrelu_kernel(
    float* __restrict__ y, const float* __restrict__ stats)
{
  const int i = blockIdx.x * 256 + threadIdx.x;
  const int o = (i >> 10) & (kC - 1);   // layout [b][o][m], m-dim = 1024
  const float v = y[i] * stats[2 * o] + stats[2 * o + 1];
  y[i] = (v >= 0.f) ? v : 0.2f * v;
}

// ---------------------------------------------------------------------------
extern "C" void kernel_launch(void* const* d_in, const int* in_sizes, int n_in,
                              void* d_out, int out_size, void* d_ws, size_t ws_size,
                              hipStream_t stream)
{
  const float* weight      = (const float*)d_in[0];   // [B,N]
  const float* allfeature  = (const float*)d_in[1];   // [B,N,C]
  const float* keyfeature  = (const float*)d_in[2];   // [B,KK,C]
  const float* refinepoint = (const float*)d_in[3];   // [B,N,3]
  // d_in[4] = keypoint (unused by reference forward)
  const int*   topidx      = (const int*)d_in[5];     // [B,KK] int32
  // d_in[6] = k (compile-time kKNN = 20)
  const float* conv_w      = (const float*)d_in[7];   // [C,C]
  const float* conv_b      = (const float*)d_in[8];   // [C]
  const float* bn_gamma    = (const float*)d_in[9];   // [C]
  const float* bn_beta     = (const float*)d_in[10];  // [C]

  char* ws = (char*)d_ws;
  int*   knn_idx = (int*)ws;                              // 2048*20*4   = 163840 B
  float* feature = (float*)(ws + 163840);                 // 2048*128*4  = 1 MiB
  float* stats   = (float*)(ws + 163840 + 1048576);       // 128*2*4     = 1 KiB
  float* y       = (float*)d_out;                         // [B,C,KK] fp32

  knn_kernel     <<<kM / 8, 256, 0, stream>>>(refinepoint, topidx, knn_idx);
  feat_kernel    <<<kM, 128, 0, stream>>>(allfeature, weight, keyfeature, knn_idx, feature);
  gemm_kernel    <<<(kB * (kC / 16) * (kKK / 16)) / 4, 128, 0, stream>>>(conv_w, conv_b, feature, y);
  stats_kernel   <<<kC, 256, 0, stream>>>(y, bn_gamma, bn_beta, stats);
  bn_lrelu_kernel<<<(kB * kC * kKK) / 256, 256, 0, stream>>>(y, stats);
}